// Attention_Decoder_69389491634628
// MI455X (gfx1250) — compile-verified
//
#include <hip/hip_runtime.h>
#include <math.h>

typedef __attribute__((ext_vector_type(2))) float v2f;
typedef __attribute__((ext_vector_type(8))) float v8f;

#define B_   128
#define TE_  512
#define TD_  128
#define EH_  512
#define DH_  512
#define KC_  7
#define CC_  105
#define WK_  617        // C_CNN + EH
#define XK_  624        // WK_ padded to multiple of 16
#define G4_  2048       // 4*DH

// ---------------------------------------------------------------------------
// phi[t,e] = sum_{t'<t} phi_w[e,t'] + phi_b[e]   (exclusive prefix sum)
__global__ void k_phi(const float* __restrict__ phi_w, const float* __restrict__ phi_b,
                      float* __restrict__ phi) {
  int e = blockIdx.x * blockDim.x + threadIdx.x;
  if (e >= EH_) return;
  float acc = 0.f, pb = phi_b[e];
  for (int t = 0; t < TE_; ++t) {
    phi[t * EH_ + e] = acc + pb;
    acc += phi_w[e * TE_ + t];
  }
}

// Wa[b,i] = s_i[b,:] . Wa_w[i,:] + Wa_b[i]
__global__ void k_wa(const float* __restrict__ s_i, const float* __restrict__ Wa_w,
                     const float* __restrict__ Wa_b, float* __restrict__ Wa) {
  int id = blockIdx.x * blockDim.x + threadIdx.x;
  if (id >= B_ * TD_) return;
  int b = id / TD_, i = id % TD_;
  float acc = Wa_b[i];
  const float* s = s_i + b * DH_;
  const float* w = Wa_w + i * DH_;
  for (int d = 0; d < DH_; ++d) acc += s[d] * w[d];
  Wa[id] = acc;
}

// r[b,g] = bih[g] + bhh[g] + s_i[b,:] . Whh[g,:]
__global__ void k_r(const float* __restrict__ s_i, const float* __restrict__ Whh,
                    const float* __restrict__ bih, const float* __restrict__ bhh,
                    float* __restrict__ r) {
  int id = blockIdx.x * blockDim.x + threadIdx.x;
  if (id >= B_ * G4_) return;
  int b = id / G4_, g = id % G4_;
  float acc = bih[g] + bhh[g];
  const float* s = s_i + b * DH_;
  const float* w = Whh + (size_t)g * DH_;
  for (int d = 0; d < DH_; ++d) acc += s[d] * w[d];
  r[id] = acc;
}

// ep[b,k] = tanh( sum_c CNNs[b,k,c]*wij_w[k,c] + s_i[b,:] . wij_w[k,105:] + wij_b[k] )
__global__ void k_ep(const float* __restrict__ CNNs, const float* __restrict__ wij_w,
                     const float* __restrict__ wij_b, const float* __restrict__ s_i,
                     float* __restrict__ ep) {
  int id = blockIdx.x * blockDim.x + threadIdx.x;
  if (id >= B_ * KC_) return;
  int b = id / KC_, k = id % KC_;
  float acc = wij_b[k];
  const float* cn = CNNs + (size_t)id * CC_;
  const float* w  = wij_w + (size_t)k * WK_;
  for (int c = 0; c < CC_; ++c) acc += cn[c] * w[c];
  const float* s  = s_i + b * DH_;
  const float* ws = w + CC_;
  for (int d = 0; d < DH_; ++d) acc += s[d] * ws[d];
  ep[id] = tanhf(acc);
}

// ap = exp(ep)/sum(ep)  (GLOBAL sum of raw ep, per reference);  c_p[b,c]=sum_k ap*CNNs
__global__ void k_cp(const float* __restrict__ ep, const float* __restrict__ CNNs,
                     float* __restrict__ cp) {
  __shared__ float red[256];
  int tid = threadIdx.x;
  float s = 0.f;
  for (int n = tid; n < B_ * KC_; n += 256) s += ep[n];
  red[tid] = s;
  __syncthreads();
  for (int off = 128; off > 0; off >>= 1) {
    if (tid < off) red[tid] += red[tid + off];
    __syncthreads();
  }
  float invS = 1.f / red[0];
  for (int idx = tid; idx < B_ * CC_; idx += 256) {
    int b = idx / CC_, c = idx % CC_;
    float acc = 0.f;
    for (int k = 0; k < KC_; ++k)
      acc += expf(ep[b * KC_ + k]) * invS * CNNs[(size_t)(b * KC_ + k) * CC_ + c];
    cp[idx] = acc;
  }
}

// Wih (2048x617) -> zero-padded (2048x624)
__global__ void k_padw(const float* __restrict__ Wih, float* __restrict__ Wp) {
  int id = blockIdx.x * blockDim.x + threadIdx.x;
  if (id >= G4_ * XK_) return;
  int g = id / XK_, k = id % XK_;
  Wp[id] = (k < WK_) ? Wih[(size_t)g * WK_ + k] : 0.f;
}

// x[b,i,512:624] = c_p broadcast (cols 512..616) / zero pad (617..623)
__global__ void k_fillx(const float* __restrict__ cp, float* __restrict__ xpad) {
  int id = blockIdx.x * blockDim.x + threadIdx.x;
  if (id >= B_ * TD_ * (XK_ - EH_)) return;
  int c = id % (XK_ - EH_);
  int bi = id / (XK_ - EH_);
  int b = bi / TD_;
  xpad[(size_t)bi * XK_ + EH_ + c] = (c < CC_) ? cp[b * CC_ + c] : 0.f;
}

// ---------------------------------------------------------------------------
// WMMA #1: banded Ua GEMM (2x2 register-blocked, 32x32 per wave),
// fused ph = phi*LSTM on the A side, epilogue:
// E[b,i,t] = tanh(Ua[b,t,s] + Ua_b[s]) * va_w[s+128] + va_b,  s = i+384-t
__global__ void __launch_bounds__(32) k_uband(
    const float* __restrict__ LSTM, const float* __restrict__ phi,
    const float* __restrict__ Ua_w, const float* __restrict__ Ua_b,
    const float* __restrict__ va_w, const float* __restrict__ va_b,
    float* __restrict__ E) {
  int Ti = blockIdx.x;                    // 0..15, 32-row t-block
  int Si = 11 - Ti + (int)blockIdx.y;     // diagonal band: Ti+Si in [11,15]
  int b  = blockIdx.z;
  if (Si < 0 || Si > 15) return;
  int t0 = Ti * 32, s0 = Si * 32;
  int lane = threadIdx.x;
  int hl = lane >> 4, l15 = lane & 15;

  v8f acc[2][2];
#pragma unroll
  for (int ni = 0; ni < 2; ++ni) {
    float ub = Ua_b[s0 + 16 * ni + l15];  // bias depends only on column s
#pragma unroll
    for (int mi = 0; mi < 2; ++mi)
#pragma unroll
      for (int j = 0; j < 8; ++j) acc[mi][ni][j] = ub;
  }

  const float* A0 = LSTM + (size_t)(b * TE_ + t0 + l15) * EH_;
  const float* A1 = A0 + 16 * EH_;
  const float* P0 = phi + (size_t)(t0 + l15) * EH_;
  const float* P1 = P0 + 16 * EH_;
  const float* B0 = Ua_w + (size_t)(s0 + l15) * EH_;
  const float* B1 = B0 + 16 * EH_;

  for (int k0 = 0; k0 < EH_; k0 += 4) {
    int kk = k0 + 2 * hl;
    v2f a0 = *(const v2f*)(A0 + kk) * *(const v2f*)(P0 + kk);
    v2f a1 = *(const v2f*)(A1 + kk) * *(const v2f*)(P1 + kk);
    v2f b0 = *(const v2f*)(B0 + kk);
    v2f b1 = *(const v2f*)(B1 + kk);
    acc[0][0] = __builtin_amdgcn_wmma_f32_16x16x4_f32(false, a0, false, b0,
                                                      (short)0, acc[0][0], false, false);
    acc[0][1] = __builtin_amdgcn_wmma_f32_16x16x4_f32(false, a0, false, b1,
                                                      (short)0, acc[0][1], false, false);
    acc[1][0] = __builtin_amdgcn_wmma_f32_16x16x4_f32(false, a1, false, b0,
                                                      (short)0, acc[1][0], false, false);
    acc[1][1] = __builtin_amdgcn_wmma_f32_16x16x4_f32(false, a1, false, b1,
                                                      (short)0, acc[1][1], false, false);
  }

  float vb = va_b[0];
#pragma unroll
  for (int ni = 0; ni < 2; ++ni) {
    int s = s0 + 16 * ni + l15;
    float vw = va_w[s + TD_];
#pragma unroll
    for (int mi = 0; mi < 2; ++mi) {
#pragma unroll
      for (int j = 0; j < 8; ++j) {
        int t = t0 + 16 * mi + j + 8 * hl;
        int i = s + t - (TE_ - TD_);      // i = s + t - 384
        if (i >= 0 && i < TD_)
          E[(size_t)(b * TD_ + i) * TE_ + t] = tanhf(acc[mi][ni][j]) * vw + vb;
      }
    }
  }
}

// softmax over t for each (b,i); t > i+384 uses the Wa branch; in-place -> a
__global__ void k_softmax(float* __restrict__ E, const float* __restrict__ Wa,
                          const float* __restrict__ va_w, const float* __restrict__ va_b) {
  int row = blockIdx.x * (blockDim.x >> 5) + (threadIdx.x >> 5);
  int lane = threadIdx.x & 31;
  if (row >= B_ * TD_) return;
  int b = row / TD_, i = row % TD_;
  float* Er = E + (size_t)row * TE_;
  float vb = va_b[0];
  float v[16];
  float mx = -1e30f;
#pragma unroll
  for (int k = 0; k < 16; ++k) {
    int t = lane + 32 * k;
    float x;
    if (t <= i + (TE_ - TD_)) {
      x = Er[t];
    } else {
      int j = i + TE_ - t;                     // 1..127
      x = tanhf(Wa[b * TD_ + j]) * va_w[j] + vb;
    }
    v[k] = x;
    mx = fmaxf(mx, x);
  }
  for (int off = 16; off > 0; off >>= 1) mx = fmaxf(mx, __shfl_xor(mx, off, 32));
  float sum = 0.f;
#pragma unroll
  for (int k = 0; k < 16; ++k) { v[k] = expf(v[k] - mx); sum += v[k]; }
  for (int off = 16; off > 0; off >>= 1) sum += __shfl_xor(sum, off, 32);
  float inv = 1.f / sum;
#pragma unroll
  for (int k = 0; k < 16; ++k) Er[lane + 32 * k] = v[k] * inv;
}

// WMMA #2: c_i[b,i,e] = sum_t a[b,i,t]*LSTM[b,t,e] -> xpad[:, :, 0:512]
// 2x2 register-blocked (32x32 per wave)
__global__ void __launch_bounds__(32) k_ci(
    const float* __restrict__ A, const float* __restrict__ LSTM,
    float* __restrict__ xpad) {
  int tile = blockIdx.x;                  // 4 m-blocks x 16 n-blocks
  int b = blockIdx.y;
  int i0 = (tile >> 4) * 32, n0 = (tile & 15) * 32;
  int lane = threadIdx.x, hl = lane >> 4, l15 = lane & 15;

  v8f acc[2][2] = {};
  const float* A0 = A + (size_t)(b * TD_ + i0 + l15) * TE_;
  const float* A1 = A0 + 16 * TE_;
  const float* Bb0 = LSTM + (size_t)b * TE_ * EH_ + (n0 + l15);
  const float* Bb1 = Bb0 + 16;

  for (int k0 = 0; k0 < TE_; k0 += 4) {
    int kk = k0 + 2 * hl;
    v2f a0 = *(const v2f*)(A0 + kk);
    v2f a1 = *(const v2f*)(A1 + kk);
    v2f b0, b1;
    b0.x = Bb0[(size_t)kk * EH_];
    b0.y = Bb0[(size_t)(kk + 1) * EH_];
    b1.x = Bb1[(size_t)kk * EH_];
    b1.y = Bb1[(size_t)(kk + 1) * EH_];
    acc[0][0] = __builtin_amdgcn_wmma_f32_16x16x4_f32(false, a0, false, b0,
                                                      (short)0, acc[0][0], false, false);
    acc[0][1] = __builtin_amdgcn_wmma_f32_16x16x4_f32(false, a0, false, b1,
                                                      (short)0, acc[0][1], false, false);
    acc[1][0] = __builtin_amdgcn_wmma_f32_16x16x4_f32(false, a1, false, b0,
                                                      (short)0, acc[1][0], false, false);
    acc[1][1] = __builtin_amdgcn_wmma_f32_16x16x4_f32(false, a1, false, b1,
                                                      (short)0, acc[1][1], false, false);
  }
#pragma unroll
  for (int mi = 0; mi < 2; ++mi)
#pragma unroll
    for (int ni = 0; ni < 2; ++ni)
#pragma unroll
      for (int j = 0; j < 8; ++j) {
        int m = 16 * mi + j + 8 * hl;
        xpad[(size_t)(b * TD_ + i0 + m) * XK_ + n0 + 16 * ni + l15] = acc[mi][ni][j];
      }
}

// WMMA #3: 2 M-tiles x 4 gate tiles share A fragments; fused LSTM-cell -> h
__global__ void __launch_bounds__(32) k_gates(
    const float* __restrict__ xpad, const float* __restrict__ Wp,
    const float* __restrict__ r, const float* __restrict__ m_i,
    float* __restrict__ out) {
  int tile = blockIdx.x;                  // 4 m-blocks x 32 d-tiles
  int b = blockIdx.y;
  int i0 = (tile >> 5) * 32, d0 = (tile & 31) * 16;
  int lane = threadIdx.x, hl = lane >> 4, l15 = lane & 15;

  v8f acc[2][4];
  const float* rb = r + (size_t)b * G4_;
#pragma unroll
  for (int g = 0; g < 4; ++g) {
    float rv = rb[g * DH_ + d0 + l15];    // bias+recurrent, col-only
#pragma unroll
    for (int mi = 0; mi < 2; ++mi)
#pragma unroll
      for (int j = 0; j < 8; ++j) acc[mi][g][j] = rv;
  }

  const float* A0 = xpad + (size_t)(b * TD_ + i0 + l15) * XK_;
  const float* A1 = A0 + 16 * XK_;
  for (int k0 = 0; k0 < XK_; k0 += 4) {
    int kk = k0 + 2 * hl;
    v2f a0 = *(const v2f*)(A0 + kk);
    v2f a1 = *(const v2f*)(A1 + kk);
#pragma unroll
    for (int g = 0; g < 4; ++g) {
      v2f bf = *(const v2f*)(Wp + (size_t)(g * DH_ + d0 + l15) * XK_ + kk);
      acc[0][g] = __builtin_amdgcn_wmma_f32_16x16x4_f32(false, a0, false, bf,
                                                        (short)0, acc[0][g], false, false);
      acc[1][g] = __builtin_amdgcn_wmma_f32_16x16x4_f32(false, a1, false, bf,
                                                        (short)0, acc[1][g], false, false);
    }
  }

  float mv = m_i[(size_t)b * DH_ + d0 + l15];
#pragma unroll
  for (int mi = 0; mi < 2; ++mi) {
#pragma unroll
    for (int j = 0; j < 8; ++j) {
      int m = 16 * mi + j + 8 * hl;
      float ig = acc[mi][0][j], fg = acc[mi][1][j];
      float gg = acc[mi][2][j], og = acc[mi][3][j];
      float si = 1.f / (1.f + expf(-ig));
      float sf = 1.f / (1.f + expf(-fg));
      float so = 1.f / (1.f + expf(-og));
      float c = sf * mv + si * tanhf(gg);
      out[(size_t)(b * TD_ + i0 + m) * DH_ + d0 + l15] = so * tanhf(c);
    }
  }
}

// ---------------------------------------------------------------------------
extern "C" void kernel_launch(void* const* d_in, const int* in_sizes, int n_in,
                              void* d_out, int out_size, void* d_ws, size_t ws_size,
                              hipStream_t stream) {
  (void)in_sizes; (void)n_in; (void)out_size; (void)ws_size;
  const float* LSTM  = (const float*)d_in[0];
  const float* CNNs  = (const float*)d_in[1];
  const float* Wa_w  = (const float*)d_in[2];
  const float* Wa_b  = (const float*)d_in[3];
  const float* Ua_w  = (const float*)d_in[4];
  const float* Ua_b  = (const float*)d_in[5];
  const float* va_w  = (const float*)d_in[6];
  const float* va_b  = (const float*)d_in[7];
  const float* phi_w = (const float*)d_in[8];
  const float* phi_b = (const float*)d_in[9];
  const float* Wih   = (const float*)d_in[10];
  const float* Whh   = (const float*)d_in[11];
  const float* bih   = (const float*)d_in[12];
  const float* bhh   = (const float*)d_in[13];
  const float* wij_w = (const float*)d_in[14];
  const float* wij_b = (const float*)d_in[15];
  const float* s_i   = (const float*)d_in[16];
  const float* m_i   = (const float*)d_in[17];
  float* out = (float*)d_out;

  float* ws   = (float*)d_ws;
  float* phi  = ws;                                   // 512*512
  float* Wa   = phi  + TE_ * EH_;                     // 128*128
  float* r    = Wa   + B_ * TD_;                      // 128*2048
  float* ep   = r    + B_ * G4_;                      // 128*7
  float* cp   = ep   + B_ * KC_;                      // 128*105
  float* E    = cp   + B_ * CC_ + 15;                 // align pad
  E = (float*)(((uintptr_t)E) & ~(uintptr_t)63);      // 64B align
  float* xpad = E    + (size_t)B_ * TD_ * TE_;        // 128*128*512
  float* Wp   = xpad + (size_t)B_ * TD_ * XK_;        // 128*128*624 ; Wp: 2048*624

  k_phi  <<<dim3(2), dim3(256), 0, stream>>>(phi_w, phi_b, phi);
  k_wa   <<<dim3(64), dim3(256), 0, stream>>>(s_i, Wa_w, Wa_b, Wa);
  k_r    <<<dim3(1024), dim3(256), 0, stream>>>(s_i, Whh, bih, bhh, r);
  k_ep   <<<dim3(4), dim3(256), 0, stream>>>(CNNs, wij_w, wij_b, s_i, ep);
  k_cp   <<<dim3(1), dim3(256), 0, stream>>>(ep, CNNs, cp);
  k_padw <<<dim3((G4_ * XK_ + 255) / 256), dim3(256), 0, stream>>>(Wih, Wp);

  k_uband<<<dim3(16, 5, B_), dim3(32), 0, stream>>>(LSTM, phi, Ua_w, Ua_b, va_w, va_b, E);
  k_softmax<<<dim3(B_ * TD_ / 8), dim3(256), 0, stream>>>(E, Wa, va_w, va_b);
  k_ci   <<<dim3(64, B_), dim3(32), 0, stream>>>(E, LSTM, xpad);
  k_fillx<<<dim3((B_ * TD_ * (XK_ - EH_) + 255) / 256), dim3(256), 0, stream>>>(cp, xpad);
  k_gates<<<dim3(128, B_), dim3(32), 0, stream>>>(xpad, Wp, r, m_i, out);
}